// AttentionPooling_67302137528777
// MI455X (gfx1250) — compile-verified
//
#include <hip/hip_runtime.h>
#include <hip/hip_bf16.h>

// AttentionPooling: B=64, S=4096, D=768 (fp32 in, fp32 out).
// Single-pass online-softmax pooled attention, split-K over the sequence.
//   Kernel 1: 512 blocks (8 splits x 64 batches), each streams 512 tokens once,
//             fused LN+score, online softmax, WMMA f32 16x16x4 pooled accumulate.
//   Kernel 2: 64 blocks merge the 8 split partials and normalize.
// Workspace requirement: 64*8*770*4 = 1,576,960 bytes of d_ws.

typedef float v8f __attribute__((ext_vector_type(8)));
typedef float v2f __attribute__((ext_vector_type(2)));

static constexpr int Bsz      = 64;
static constexpr int Ssz      = 4096;
static constexpr int Dsz      = 768;
static constexpr int SPLIT    = 8;
static constexpr int TOKS     = Ssz / SPLIT;   // 512 tokens per block
static constexpr int CT       = 16;            // tokens per chunk
static constexpr int NCHUNK   = TOKS / CT;     // 32
static constexpr int LDST     = 776;           // LDS row stride (floats); 776 % 32 == 8
static constexpr int LDST4    = LDST / 4;      // 194 float4
static constexpr int THREADS  = 256;
static constexpr int NW       = THREADS / 32;  // 8 waves
static constexpr int NTILE    = Dsz / 16 / NW; // 6 d-tiles per wave
static constexpr int WS_STRIDE = 2 + Dsz;      // m, l, acc[768]
static constexpr float NEG_BIG = -3.402823466e38f;

__global__ __launch_bounds__(THREADS)
void attnpool_partial(const float* __restrict__ X,
                      const float* __restrict__ w,
                      const float* __restrict__ gamma,
                      const float* __restrict__ beta,
                      float* __restrict__ ws) {
  __shared__ float Xl[CT * LDST];   // staged token chunk (~48.5 KB)
  __shared__ float u_lds[Dsz];      // gamma * w
  __shared__ float sc[CT];          // chunk scores
  __shared__ float pbuf[CT];        // chunk exp-weights

  const int tid   = threadIdx.x;
  const int lane  = tid & 31;
  const int wave  = tid >> 5;
  const int half  = lane >> 4;      // 0: lanes 0-15, 1: lanes 16-31
  const int mrow  = lane & 15;
  const int split = blockIdx.x;
  const int b     = blockIdx.y;

  // staging assignment: thread owns LDS row `trow`, columns crow, crow+16, ... (float4 units)
  const int trow = tid >> 4;        // 0..15 (token within chunk)
  const int crow = tid & 15;        // 0..15 (float4 column base)

  // ---- prologue: u = gamma.*w in LDS; per-wave redundant reduce of g=sum(u), bw=sum(beta.*w)
  for (int d = tid; d < Dsz; d += THREADS) u_lds[d] = gamma[d] * w[d];
  __syncthreads();
  float gg = 0.f, bw = 0.f;
  for (int d = lane; d < Dsz; d += 32) {
    gg += u_lds[d];
    bw = fmaf(beta[d], w[d], bw);
  }
  for (int off = 16; off; off >>= 1) {
    gg += __shfl_xor(gg, off, 32);
    bw += __shfl_xor(bw, off, 32);
  }

  float m_run = NEG_BIG, l_run = 0.f;
  v8f cacc[NTILE];
  for (int i = 0; i < NTILE; ++i)
    for (int e = 0; e < 8; ++e) cacc[i][e] = 0.f;

  for (int c = 0; c < NCHUNK; ++c) {
    __syncthreads();  // protect Xl/pbuf reuse from previous chunk
    const int s0 = split * TOKS + c * CT;

    // ---- stage chunk (16 contiguous tokens = 48 KiB) into LDS.
    // Thread loads 12 float4s of its own row: pure constant-offset addressing.
    {
      const float4* g4 =
          reinterpret_cast<const float4*>(X + ((size_t)b * Ssz + s0) * Dsz) +
          (size_t)trow * (Dsz / 4) + crow;
      float4* l4 = reinterpret_cast<float4*>(Xl) + trow * LDST4 + crow;
#pragma unroll
      for (int k = 0; k < (Dsz / 4) / 16; ++k) {  // 12
        l4[16 * k] = g4[16 * k];
      }
    }
    __syncthreads();

    // ---- fused LayerNorm + score: each wave handles tokens {wave, wave+8}
    for (int tt = 0; tt < CT / NW; ++tt) {
      const int t = wave + NW * tt;
      const float* row = &Xl[t * LDST];
      float s1 = 0.f, s2 = 0.f, su = 0.f;
      for (int d = lane; d < Dsz; d += 32) {
        float x = row[d];
        s1 += x;
        s2 = fmaf(x, x, s2);
        su = fmaf(u_lds[d], x, su);
      }
      for (int off = 16; off; off >>= 1) {
        s1 += __shfl_xor(s1, off, 32);
        s2 += __shfl_xor(s2, off, 32);
        su += __shfl_xor(su, off, 32);
      }
      float mu   = s1 * (1.f / Dsz);
      float var  = s2 * (1.f / Dsz) - mu * mu;
      float rstd = rsqrtf(var + 1e-5f);
      if (lane == 0) sc[t] = rstd * (su - mu * gg) + bw;
    }
    __syncthreads();

    // ---- online softmax bookkeeping (redundant per-thread, deterministic)
    float cmax = NEG_BIG;
    for (int t = 0; t < CT; ++t) cmax = fmaxf(cmax, sc[t]);
    float m_new = fmaxf(m_run, cmax);
    float scale = __expf(m_run - m_new);
    if (tid < CT) pbuf[tid] = __expf(sc[tid] - m_new);
    __syncthreads();
    float ps = 0.f;
    for (int t = 0; t < CT; ++t) ps += pbuf[t];
    l_run = fmaf(l_run, scale, ps);
    m_run = m_new;

    // Rescale pooled accumulators. A-rows m>0 are zero and C starts at zero,
    // so only element [0] (C row 0 in lanes 0-15) ever carries data.
    for (int i = 0; i < NTILE; ++i) cacc[i][0] *= scale;

    // ---- WMMA pooled accumulate: C[0,n] += sum_k p[k0+k] * X[s0+k0+k, dbase+n]
    // A (16x4 f32): row m=0 holds p; lanes0-15 -> K0,K1; lanes16-31 -> K2,K3.
    // B (4x16 f32): lanes0-15 n=lane rows K0(v0),K1(v1); lanes16-31 rows K2,K3.
    for (int kg = 0; kg < CT / 4; ++kg) {
      const int k0 = kg * 4;
      const int ra = k0 + 2 * half;
      v2f a;
      a.x = (mrow == 0) ? pbuf[ra]     : 0.f;
      a.y = (mrow == 0) ? pbuf[ra + 1] : 0.f;
#pragma unroll
      for (int i = 0; i < NTILE; ++i) {
        const int dbase = (wave + NW * i) * 16;
        v2f bf;
        bf.x = Xl[ra * LDST + dbase + mrow];
        bf.y = Xl[(ra + 1) * LDST + dbase + mrow];
        cacc[i] = __builtin_amdgcn_wmma_f32_16x16x4_f32(
            false, a, false, bf, (short)0, cacc[i], false, false);
      }
    }
  }

  // ---- write split partials: [m, l, acc[768]]
  const size_t base = ((size_t)b * SPLIT + split) * WS_STRIDE;
  if (tid == 0) { ws[base] = m_run; ws[base + 1] = l_run; }
  for (int i = 0; i < NTILE; ++i) {
    const int dbase = (wave + NW * i) * 16;
    if (lane < 16) ws[base + 2 + dbase + lane] = cacc[i][0];  // C row 0, N=lane
  }
}

__global__ __launch_bounds__(THREADS)
void attnpool_combine(const float* __restrict__ ws, float* __restrict__ out) {
  const int b = blockIdx.x;
  const int tid = threadIdx.x;
  float mj[SPLIT], lj[SPLIT], coef[SPLIT];
  float mg = NEG_BIG;
  for (int j = 0; j < SPLIT; ++j) {
    const size_t base = ((size_t)b * SPLIT + j) * WS_STRIDE;
    mj[j] = ws[base];
    lj[j] = ws[base + 1];
    mg = fmaxf(mg, mj[j]);
  }
  float L = 0.f;
  for (int j = 0; j < SPLIT; ++j) {
    coef[j] = __expf(mj[j] - mg);
    L = fmaf(lj[j], coef[j], L);
  }
  const float inv = 1.f / L;
  for (int d = tid; d < Dsz; d += THREADS) {
    float s = 0.f;
    for (int j = 0; j < SPLIT; ++j) {
      const size_t base = ((size_t)b * SPLIT + j) * WS_STRIDE;
      s = fmaf(coef[j], ws[base + 2 + d], s);
    }
    out[(size_t)b * Dsz + d] = s * inv;
  }
}

extern "C" void kernel_launch(void* const* d_in, const int* in_sizes, int n_in,
                              void* d_out, int out_size, void* d_ws, size_t ws_size,
                              hipStream_t stream) {
  const float* X     = (const float*)d_in[0];  // [B,S,D]
  const float* w     = (const float*)d_in[1];  // [D]
  const float* gamma = (const float*)d_in[2];  // [D]
  const float* beta  = (const float*)d_in[3];  // [D]
  float* out = (float*)d_out;                  // [B,D]
  float* ws  = (float*)d_ws;                   // needs 64*8*770 floats

  dim3 grid1(SPLIT, Bsz);
  attnpool_partial<<<grid1, THREADS, 0, stream>>>(X, w, gamma, beta, ws);
  attnpool_combine<<<Bsz, THREADS, 0, stream>>>(ws, out);
}